// ELR_43241730736271
// MI455X (gfx1250) — compile-verified
//
#include <hip/hip_runtime.h>
#include <math.h>

typedef float v4f __attribute__((ext_vector_type(4)));
typedef int   v4i_vs __attribute__((__vector_size__(4 * sizeof(int))));   // builtin's element type
typedef __attribute__((address_space(3))) v4i_vs lds_v4i;                  // AS3 (LDS) variant

#define C_CLASSES 100
#define LANES_ACT 25          // 100 floats / 4 per lane = 25 active lanes (wave32)
#define WAVES_PER_BLOCK 8     // 256 threads = 8 wave32s

#ifndef __has_builtin
#define __has_builtin(x) 0
#endif

// ---- CDNA5 async global->LDS path (ASYNCcnt-tracked), probe via __has_builtin ----
#if __has_builtin(__builtin_amdgcn_global_load_async_to_lds_b128)
#define HAVE_ASYNC_LDS 1
#else
#define HAVE_ASYNC_LDS 0
#endif

#if __has_builtin(__builtin_amdgcn_s_wait_asynccnt)
#define WAIT_ASYNC0() __builtin_amdgcn_s_wait_asynccnt(0)
#else
#define WAIT_ASYNC0() asm volatile("s_wait_asynccnt 0" ::: "memory")
#endif

__device__ __forceinline__ float wred_max32(float v) {
  for (int o = 16; o; o >>= 1) v = fmaxf(v, __shfl_xor(v, o, 32));
  return v;
}
__device__ __forceinline__ float wred_sum32(float v) {
  for (int o = 16; o; o >>= 1) v += __shfl_xor(v, o, 32);
  return v;
}

// ---------------- kernel 0: zero the double loss accumulator ----------------
__global__ void ELR_zeroAcc(double* acc) { acc[0] = 0.0; }

// ---------------- kernel 1: stream-copy target -> new_target ----------------
// dst = d_out + 1  (only 4B-aligned) -> B128 NT loads from src, B32 NT stores.
__global__ __launch_bounds__(256) void ELR_copy_tgt(const float* __restrict__ src,
                                                    float* __restrict__ dst,
                                                    unsigned long long n) {
  const unsigned long long n4 = n >> 2;
  const unsigned long long stride = (unsigned long long)gridDim.x * blockDim.x;
  const v4f* __restrict__ s4 = (const v4f*)src;
  unsigned long long gid = (unsigned long long)blockIdx.x * blockDim.x + threadIdx.x;
  for (unsigned long long i = gid; i < n4; i += stride) {
    v4f v = __builtin_nontemporal_load(&s4[i]);
    float* d = dst + (i << 2);
    __builtin_nontemporal_store(v.x, d + 0);
    __builtin_nontemporal_store(v.y, d + 1);
    __builtin_nontemporal_store(v.z, d + 2);
    __builtin_nontemporal_store(v.w, d + 3);
  }
  const unsigned long long tail = n4 << 2;
  if (gid < n - tail) dst[tail + gid] = src[tail + gid];
}

// ---------------- kernel 2: per-row softmax/EMA/loss, one wave per row ----------------
__global__ __launch_bounds__(256) void ELR_rows(const float* __restrict__ logits,
                                                const float* __restrict__ target,
                                                const int* __restrict__ label,
                                                const int* __restrict__ index,
                                                float* __restrict__ newTgt,
                                                double* __restrict__ lossAcc,
                                                int B) {
  const int w    = threadIdx.x >> 5;
  const int lane = threadIdx.x & 31;
  const int row  = blockIdx.x * WAVES_PER_BLOCK + w;
  float rowLoss = 0.0f;

#if HAVE_ASYNC_LDS
  __shared__ v4f stageX[WAVES_PER_BLOCK][LANES_ACT];
  __shared__ v4f stageT[WAVES_PER_BLOCK][LANES_ACT];
#endif

  if (row < B) {
    const bool act = lane < LANES_ACT;
    const int idx = index[row];
    const int lab = label[row];
    const v4f* __restrict__ xrow = (const v4f*)(logits + (unsigned long long)row * C_CLASSES);
    const v4f* __restrict__ trow = (const v4f*)(target + (unsigned long long)idx * C_CLASSES);

    v4f x = {-INFINITY, -INFINITY, -INFINITY, -INFINITY};
    v4f t = {0.0f, 0.0f, 0.0f, 0.0f};
#if HAVE_ASYNC_LDS
    if (act) {
      __builtin_amdgcn_global_load_async_to_lds_b128(
          (v4i_vs*)(xrow + lane), (lds_v4i*)&stageX[w][lane], 0, 0);
      __builtin_amdgcn_global_load_async_to_lds_b128(
          (v4i_vs*)(trow + lane), (lds_v4i*)&stageT[w][lane], 0, 0);
    }
    WAIT_ASYNC0();
    if (act) { x = stageX[w][lane]; t = stageT[w][lane]; }
#else
    if (act) { x = xrow[lane]; t = trow[lane]; }
#endif

    // ---- softmax with max-shift ----
    const float m = wred_max32(fmaxf(fmaxf(x.x, x.y), fmaxf(x.z, x.w)));
    v4f e;
    e.x = expf(x.x - m); e.y = expf(x.y - m);
    e.z = expf(x.z - m); e.w = expf(x.w - m);
    // e^10 for log_softmax(10*x): exp(10(x-m)) == (exp(x-m))^10
    v4f e2 = e * e, e4 = e2 * e2, e8 = e4 * e4, e10 = e8 * e2;

    float se  = e.x + e.y + e.z + e.w;
    float s10 = e10.x + e10.y + e10.z + e10.w;
    const int j0 = lane * 4;
    float xl = 0.0f;
    xl += (j0 + 0 == lab) ? x.x : 0.0f;
    xl += (j0 + 1 == lab) ? x.y : 0.0f;
    xl += (j0 + 2 == lab) ? x.z : 0.0f;
    xl += (j0 + 3 == lab) ? x.w : 0.0f;
    for (int o = 16; o; o >>= 1) {      // fused 3-way butterfly all-reduce
      se  += __shfl_xor(se,  o, 32);
      s10 += __shfl_xor(s10, o, 32);
      xl  += __shfl_xor(xl,  o, 32);
    }

    const float EPS = 1e-4f, HI = 1.0f - 1e-4f;
    const float invS = 1.0f / se;
    v4f p = {0.0f, 0.0f, 0.0f, 0.0f};
    if (act) {
      p.x = fminf(fmaxf(e.x * invS, EPS), HI);
      p.y = fminf(fmaxf(e.y * invS, EPS), HI);
      p.z = fminf(fmaxf(e.z * invS, EPS), HI);
      p.w = fminf(fmaxf(e.w * invS, EPS), HI);
    }
    const float S2 = wred_sum32(p.x + p.y + p.z + p.w);

    // EMA target row: 0.7*old + 0.3*(p / S2)
    const v4f tn = 0.7f * t + (0.3f / S2) * p;
    const float dot = wred_sum32(tn.x * p.x + tn.y * p.y + tn.z * p.z + tn.w * p.w);

    if (act) {  // dst region is 4B-aligned only (loss occupies d_out[0]) -> b32 stores
      float* d = newTgt + (unsigned long long)idx * C_CLASSES + j0;
      d[0] = tn.x; d[1] = tn.y; d[2] = tn.z; d[3] = tn.w;
    }

    const float ce = 10.0f * m + logf(s10) - 10.0f * xl;  // -log_softmax(10x)[lab]
    rowLoss = ce + 3.0f * logf(1.0f - dot);
  }

  // block reduction of per-row losses (rowLoss identical on all lanes of a wave)
  __shared__ float wl[WAVES_PER_BLOCK];
  if (lane == 0) wl[w] = rowLoss;
  __syncthreads();
  if (threadIdx.x == 0) {
    float s = 0.0f;
    for (int i = 0; i < WAVES_PER_BLOCK; ++i) s += wl[i];
    atomicAdd(lossAcc, (double)s);
  }
}

// ---------------- kernel 3: finalize mean ----------------
__global__ void ELR_finalize(const double* __restrict__ acc, float* __restrict__ out, int B) {
  out[0] = (float)(acc[0] / (double)B);
}

extern "C" void kernel_launch(void* const* d_in, const int* in_sizes, int n_in,
                              void* d_out, int out_size, void* d_ws, size_t ws_size,
                              hipStream_t stream) {
  const float* logits = (const float*)d_in[0];
  const float* target = (const float*)d_in[1];
  const int*   label  = (const int*)d_in[2];
  const int*   index  = (const int*)d_in[3];
  float* out    = (float*)d_out;
  float* newTgt = out + 1;                       // [N*C] new_target region
  const int B = in_sizes[2];
  const unsigned long long nT = (unsigned long long)in_sizes[1];
  double* acc = (double*)d_ws;

  ELR_zeroAcc<<<1, 1, 0, stream>>>(acc);
  ELR_copy_tgt<<<8192, 256, 0, stream>>>(target, newTgt, nT);
  const int nBlocks = (B + WAVES_PER_BLOCK - 1) / WAVES_PER_BLOCK;
  ELR_rows<<<nBlocks, 256, 0, stream>>>(logits, target, label, index, newTgt, acc, B);
  ELR_finalize<<<1, 1, 0, stream>>>(acc, out, B);
}